// BEV2RV_76295799046937
// MI455X (gfx1250) — compile-verified
//
#include <hip/hip_runtime.h>

namespace {
constexpr int kHB = 512, kWB = 512, kN = kHB * kWB;   // BEV pixels
constexpr int kHR = 64, kWR = 2048;                   // range view
constexpr int kC = 32, kB = 2;
constexpr int kPix = 256;                             // pixels per block
constexpr unsigned kNegInfEnc = 0x007FFFFFu;          // enc(-inf)

// Order-preserving float32 -> uint32 bijection (total order, incl. -inf)
__device__ __forceinline__ unsigned enc_f32(float f) {
  unsigned u = __float_as_uint(f);
  return (u & 0x80000000u) ? ~u : (u | 0x80000000u);
}
__device__ __forceinline__ float dec_f32(unsigned u) {
  return __uint_as_float((u & 0x80000000u) ? (u & 0x7FFFFFFFu) : ~u);
}
} // namespace

#if __has_builtin(__builtin_amdgcn_global_load_async_to_lds_b128)
#define BEV2RV_HAVE_ASYNC_LDS 1
typedef int v4i __attribute__((vector_size(16)));
typedef __attribute__((address_space(1))) v4i* bev2rv_gptr;
typedef __attribute__((address_space(3))) v4i* bev2rv_lptr;
#else
#define BEV2RV_HAVE_ASYNC_LDS 0
#endif

__global__ __launch_bounds__(256) void bev2rv_init(uint4* __restrict__ out, int n4) {
  int i = blockIdx.x * blockDim.x + threadIdx.x;
  if (i < n4) out[i] = make_uint4(kNegInfEnc, kNegInfEnc, kNegInfEnc, kNegInfEnc);
}

__global__ __launch_bounds__(256) void bev2rv_scatter(
    const float* __restrict__ feat,   // [B, 32, 512, 512]
    const int*   __restrict__ zbin,   // [B, 1, 512, 512]
    unsigned*    __restrict__ out) {  // [B, 32, 64, 2048] (encoded u32)
  __shared__ float tile[kC * kPix];   // 32 KB feature tile for this block

  const int tid = threadIdx.x;
  const int blk = blockIdx.x;         // 2048 blocks total
  const int b   = blk >> 10;          // 1024 blocks per batch
  const int n0  = (blk & 1023) << 8;  // first pixel of this block

  const float* gbase = feat + (size_t)b * kC * kN + n0;

  // ---- Stage 32KB feature tile global->LDS (async DMA, overlapped with geometry) ----
  {
    const int coff = tid >> 6;        // 0..3  : channel within group of 4
    const int eoff = (tid & 63) * 4;  // 0..252: float offset (16B granules)
#pragma unroll
    for (int k = 0; k < 8; ++k) {     // 8 x (4 channels * 64 lanes * 16B) = 32KB
      const int c = k * 4 + coff;
      const float* g = gbase + (size_t)c * kN + eoff;
      float*       l = &tile[c * kPix + eoff];
#if BEV2RV_HAVE_ASYNC_LDS
      __builtin_amdgcn_global_load_async_to_lds_b128(
          (bev2rv_gptr)(void*)g,
          (bev2rv_lptr)(void*)l,
          /*offset=*/0, /*cpol=*/0);
#else
      // synchronous fallback: plain vectorized copy through VGPRs
      *(float4*)l = *(const float4*)g;
#endif
    }
  }

  // ---- Geometry (pure VALU; runs while the DMA is in flight) ----
  const int n = n0 + tid;
  const int i = n >> 9;               // BEV row (y index)
  const int j = n & 511;              // BEV col (x index)

  // np.linspace computed in f64, cast to f32
  float y = (float)( 50.0 - (double)i * (100.0 / 511.0));
  float x = (float)(-50.0 + (double)j * (100.0 / 511.0));

  // f32 op-by-op, no fma contraction (mirror JAX/numpy op order)
  float rho = sqrtf(__fadd_rn(__fmul_rn(x, x), __fmul_rn(y, y)));
  float phi = atan2f(y, x);

  const float THMAX = 0.05235987755982988f;   // radians(3)
  const float THRNG = 0.4886921905584123f;    // radians(28)

  float thl = atan2f(-1.73f, rho);
  float rlf = rintf(__fmul_rn(__fdiv_rn(__fsub_rn(THMAX, thl), THRNG), 63.0f));
  int row_low = min(63, max(0, (int)rlf));

  float colf = rintf(__fmul_rn(__fdiv_rn(__fadd_rn(phi, 3.14159274f), 6.28318548f), 2047.0f));
  int col = min(2047, max(0, (int)colf));

  // z hint: bin center, dz = 0.2, Z_MIN + dz/2 = -3.9
  float zh  = __fadd_rn(__fmul_rn((float)zbin[b * kN + n], 0.2f), -3.9f);
  float thh = atan2f(zh, rho);
  float rhf = rintf(__fmul_rn(__fdiv_rn(__fsub_rn(THMAX, thh), THRNG), 63.0f));
  int row_high = min(63, max(0, (int)rhf));

  const int rs = min(row_low, row_high);
  const int re = max(row_low, row_high);

  // ---- Wait for the tile, then scatter-max from LDS ----
#if BEV2RV_HAVE_ASYNC_LDS
#if __has_builtin(__builtin_amdgcn_s_wait_asynccnt)
  __builtin_amdgcn_s_wait_asynccnt(0);
#else
  asm volatile("s_wait_asynccnt 0x0" ::: "memory");
#endif
#endif
  __syncthreads();

  unsigned* ob = out + (size_t)b * kC * kHR * kWR + col;
#pragma unroll 4
  for (int c = 0; c < kC; ++c) {
    unsigned e = enc_f32(tile[c * kPix + tid]);   // conflict-free LDS read
    unsigned* p = ob + (size_t)(c * kHR + rs) * kWR;
    for (int r = rs; r <= re; ++r) {              // atomics resolve in 192MB L2
      atomicMax(p, e);
      p += kWR;
    }
  }
}

__global__ __launch_bounds__(256) void bev2rv_finalize(unsigned* __restrict__ io, int n4) {
  int idx = blockIdx.x * blockDim.x + threadIdx.x;
  if (idx >= n4) return;
  uint4 v = ((const uint4*)io)[idx];
  float4 f;
  f.x = (v.x == kNegInfEnc) ? 0.0f : dec_f32(v.x);
  f.y = (v.y == kNegInfEnc) ? 0.0f : dec_f32(v.y);
  f.z = (v.z == kNegInfEnc) ? 0.0f : dec_f32(v.z);
  f.w = (v.w == kNegInfEnc) ? 0.0f : dec_f32(v.w);
  ((float4*)io)[idx] = f;
}

extern "C" void kernel_launch(void* const* d_in, const int* in_sizes, int n_in,
                              void* d_out, int out_size, void* d_ws, size_t ws_size,
                              hipStream_t stream) {
  (void)in_sizes; (void)n_in; (void)d_ws; (void)ws_size;
  const float* feat = (const float*)d_in[0];
  const int*   zbin = (const int*)d_in[1];
  unsigned*    outu = (unsigned*)d_out;

  int n4 = out_size / 4;                        // out_size = 2*32*64*2048, divisible by 4
  int init_blocks = (n4 + 255) / 256;
  bev2rv_init<<<init_blocks, 256, 0, stream>>>((uint4*)outu, n4);

  int nblocks = (kB * kN) / kPix;               // 2048 blocks, 256 threads each
  bev2rv_scatter<<<nblocks, kPix, 0, stream>>>(feat, zbin, outu);

  bev2rv_finalize<<<init_blocks, 256, 0, stream>>>(outu, n4);
}